// bicon_loss_77206332112951
// MI455X (gfx1250) — compile-verified
//
#include <hip/hip_runtime.h>

// Problem constants (from reference: B=16, C=8, H=512, W=512)
#define BB 16
#define CC 8
#define HH 512
#define WW 512
#define HWSZ (HH * WW)          // 262144
#define CHW (CC * HWSZ)         // 2097152
#define TH 16
#define TW 32
#define TILES_W (WW / TW)       // 16
#define TILES_H (HH / TH)       // 32
#define NBLK (TILES_W * TILES_H * BB)  // 8192

typedef __attribute__((ext_vector_type(2))) float v2f;
typedef __attribute__((ext_vector_type(8))) float v8f;

// -------- main fused kernel: sigmoid + stencil products + 3 BCE sums --------
// One block = one (b, 16x32) tile. LDS caches sigma and log-sigma with 1-px halo
// so each sigmoid/log is computed exactly once and shared by the 8 neighbor uses.
__global__ __launch_bounds__(256) void bicon_main_kernel(
    const float* __restrict__ cmap,    // (B,8,H,W)
    const float* __restrict__ target,  // (B,1,H,W)
    const float* __restrict__ con,     // (B,8,H,W)
    float* __restrict__ partial)       // NBLK partial sums
{
    __shared__ float s_sig[CC][TH + 2][TW + 2];  // sigmoid(x); 0 outside image
    __shared__ float s_lsg[CC][TH + 2][TW + 2];  // log sigmoid(x); -1e30 outside
    __shared__ float s_t[CC][TH][TW];            // con_target tile
    __shared__ float s_red[256];

    const int blk = blockIdx.x;
    const int twi = blk % TILES_W;
    const int thi = (blk / TILES_W) % TILES_H;
    const int b   = blk / (TILES_W * TILES_H);
    const int h0 = thi * TH;
    const int w0 = twi * TW;
    const int tid = threadIdx.x;
    const size_t baseb = (size_t)b * CHW;

    float acc = 0.0f;

    // ---- Stage 1: cooperative halo load; sigmoid + logs computed once ----
    // conmap BCE (weight 0.8) folded in for interior (owned) elements.
    for (int i = tid; i < CC * (TH + 2) * (TW + 2); i += 256) {
        const int c   = i / ((TH + 2) * (TW + 2));
        const int rem = i % ((TH + 2) * (TW + 2));
        const int hh  = rem / (TW + 2);
        const int ww  = rem % (TW + 2);
        const int h = h0 - 1 + hh;
        const int w = w0 - 1 + ww;
        float sig = 0.0f;
        float lp  = -1.0e30f;  // log(0) sentinel -> clipped to -100 at use
        if ((unsigned)h < (unsigned)HH && (unsigned)w < (unsigned)WW) {
            const size_t gi = baseb + (size_t)c * HWSZ + (size_t)h * WW + w;
            const float x = cmap[gi];
            const float e = __expf(-fabsf(x));          // exp(-|x|)
            const float L = log1pf(e);                  // softplus tail
            const float r = __builtin_amdgcn_rcpf(1.0f + e);
            sig = (x >= 0.0f) ? r : e * r;              // sigmoid(x)
            lp  = fminf(x, 0.0f) - L;                   // log sigmoid(x)
            // interior element of this tile -> accumulate conmap BCE here
            if (hh >= 1 && hh <= TH && ww >= 1 && ww <= TW) {
                const float t = con[gi];
                s_t[c][hh - 1][ww - 1] = t;
                const float l1 = -fmaxf(x, 0.0f) - L;   // log(1 - sigmoid(x))
                acc -= 0.8f * (t * fmaxf(lp, -100.0f) +
                               (1.0f - t) * fmaxf(l1, -100.0f));
            }
        }
        s_sig[c][hh][ww] = sig;
        s_lsg[c][hh][ww] = lp;
    }
    __syncthreads();

    // ---- Stage 2: per-pixel stencil products, bimap BCE, decouple BCE ----
    // vote order [a7,a3,a5,a1,a2,a6,a4,a8] => v_i = sigma_i(h,w) * sigma_{7-i}(h+dh,w+dw)
    const int dh8[8] = {-1, -1, -1,  0, 0,  1, 1, 1};
    const int dw8[8] = {-1,  0,  1, -1, 1, -1, 0, 1};
    const int tx = tid & 31;
    const int ty = tid >> 5;

    for (int r = 0; r < 2; ++r) {
        const int py = ty + 8 * r;          // 0..15 within tile
        const int hh = py + 1, ww = tx + 1; // LDS coords
        const int h = h0 + py, w = w0 + tx;

        float tch[CC], scf[CC], lcf[CC];
        float tsum = 0.0f;
#pragma unroll
        for (int c = 0; c < CC; ++c) {
            tch[c] = s_t[c][py][tx];
            tsum += tch[c];
            scf[c] = s_sig[c][hh][ww];
            lcf[c] = s_lsg[c][hh][ww];
        }

        float vsum = 0.0f, vmin = 1.0e30f, bacc = 0.0f;
#pragma unroll
        for (int i = 0; i < 8; ++i) {
            const float sn = s_sig[7 - i][hh + dh8[i]][ww + dw8[i]];
            const float ln = s_lsg[7 - i][hh + dh8[i]][ww + dw8[i]];
            const float v  = scf[i] * sn;                    // a_i
            vsum += v;
            vmin = fminf(vmin, v);
            const float la  = fmaxf(lcf[i] + ln, -100.0f);   // log(a_i) clipped
            const float l1a = fmaxf(log1pf(-v), -100.0f);    // log(1-a_i) clipped
            bacc += tch[i] * la + (1.0f - tch[i]) * l1a;
        }
        acc -= 0.2f * bacc;

        const bool edge = (tsum < 8.0f) && (tsum > 0.0f);
        const float glo = vsum * 0.125f;
        const float d   = edge ? (1.0f - vmin) : glo;
        const float tt  = target[(size_t)b * HWSZ + (size_t)h * WW + w];
        const float ld  = fmaxf(__logf(d), -100.0f);
        const float l1d = fmaxf(log1pf(-d), -100.0f);
        acc -= tt * ld + (1.0f - tt) * l1d;
    }

    // ---- deterministic block reduction ----
    s_red[tid] = acc;
    __syncthreads();
    for (int s = 128; s > 0; s >>= 1) {
        if (tid < s) s_red[tid] += s_red[tid + s];
        __syncthreads();
    }
    if (tid == 0) partial[blk] = s_red[0];
}

// -------- final reduction: single wave32, tensor-core (WMMA) summation --------
// D = A(16x4) * ones(4x16) + C : each WMMA folds 64 partials into the 16x16 f32
// accumulator; total = sum of column N=0 (held by lanes 0 and 16).
__global__ __launch_bounds__(32) void bicon_final_kernel(
    const float* __restrict__ partial, float* __restrict__ out, int n)
{
    const int lane  = threadIdx.x;
    const int m     = lane & 15;
    const int kbase = (lane >> 4) * 2;  // lanes 0-15 hold K=0,1 ; lanes 16-31 K=2,3

    v8f c = {};
    v2f bv;
    bv.x = 1.0f;
    bv.y = 1.0f;

    for (int base = 0; base < n; base += 64) {
        const int i0 = base + m * 4 + kbase;
        const int i1 = i0 + 1;
        v2f a;
        a.x = (i0 < n) ? partial[i0] : 0.0f;   // select, keeps EXEC all-ones
        a.y = (i1 < n) ? partial[i1] : 0.0f;
        c = __builtin_amdgcn_wmma_f32_16x16x4_f32(
                /*neg_a=*/false, a, /*neg_b=*/false, bv,
                /*c_mod=*/(short)0, c, /*reuse_a=*/false, /*reuse_b=*/false);
    }

    // lane 0 holds D[0..7][0] in c[0..7]; lane 16 holds D[8..15][0]
    float s = c[0] + c[1] + c[2] + c[3] + c[4] + c[5] + c[6] + c[7];
    const float tot = __shfl(s, 0, 32) + __shfl(s, 16, 32);
    if (lane == 0) out[0] = tot;
}

extern "C" void kernel_launch(void* const* d_in, const int* in_sizes, int n_in,
                              void* d_out, int out_size, void* d_ws, size_t ws_size,
                              hipStream_t stream) {
    const float* cmap   = (const float*)d_in[0];  // c_map (16,8,512,512) f32
    const float* target = (const float*)d_in[1];  // target (16,1,512,512) f32
    const float* con    = (const float*)d_in[2];  // con_target (16,8,512,512) f32
    float* partial = (float*)d_ws;                // 8192 floats (32 KB)
    float* out = (float*)d_out;

    bicon_main_kernel<<<NBLK, 256, 0, stream>>>(cmap, target, con, partial);
    bicon_final_kernel<<<1, 32, 0, stream>>>(partial, out, NBLK);
}